// GNNEncoder_5291399709172
// MI455X (gfx1250) — compile-verified
//
#include <hip/hip_runtime.h>

// ---------------------------------------------------------------------------
// GINE GNN encoder for MI455X (gfx1250, wave32).
// Memory-bound problem (~2.8 GB edge-message traffic vs 32 GFLOP dense), so
// all GEMMs use V_WMMA_F32_16X16X4_F32 (full fp32 fidelity) and scatter uses
// native fp32 global atomics. A-operand fetches are 64-bit vectorized.
// ---------------------------------------------------------------------------

typedef __attribute__((ext_vector_type(2))) float v2f;
typedef __attribute__((ext_vector_type(8))) float v8f;

#define HD     128     // hidden dim
#define EIN    16      // edge input dim
#define NNODES 50000
#define NEDGES 600000
#define NGRAPH 1024
#define WAVE   32
#define WPB    4       // waves per block for GEMM kernels

__device__ __forceinline__ v8f wmma_f32(v2f a, v2f b, v8f c) {
  // D = A(16x4,f32) * B(4x16,f32) + C(16x16,f32)
  return __builtin_amdgcn_wmma_f32_16x16x4_f32(false, a, false, b, (short)0, c,
                                               false, false);
}

// C tile (16x16, column block `ncol`) = A[16xK](row-major, lda) * B[KxHD](row-major)
// + bias, accumulated over K (multiple of 4) via chained WMMA.
// A-layout (f32 16x4): lane L<16 holds row M=L, K = k+{0,1}; lanes 16-31 K = k+{2,3}.
//   -> the per-lane pair is contiguous in row-major A: load as one 64-bit v2f.
// B-layout (f32 4x16): lanes 0-15 hold N=lane with K = k+{0,1}; lanes 16-31 K = k+{2,3}.
__device__ __forceinline__ v8f gemm16(const float* __restrict__ A, int lda,
                                      const float* __restrict__ B,
                                      const float* __restrict__ bias,
                                      int ncol, int K, int lane) {
  const int row  = lane & 15;
  const int koff = (lane >> 4) << 1;
  const int col  = ncol * 16 + row;        // also bias index (per-column bias)
  const float bv = bias[col];
  v8f acc;
#pragma unroll
  for (int i = 0; i < 8; ++i) acc[i] = bv;
  for (int k = 0; k < K; k += 4) {
    v2f a = *(const v2f*)(A + row * lda + k + koff);   // 64-bit load
    v2f b;
    b.x = B[(k + koff) * HD + col];
    b.y = B[(k + koff + 1) * HD + col];
    acc = wmma_f32(a, b, acc);
  }
  return acc;
}

// D-layout (f32 16x16 C/D): VGPR r -> row M = r + 8*(lane>=16), col N = lane&15.
__device__ __forceinline__ void store_tile_lds(v8f acc, float* __restrict__ lds,
                                               int ncol, int lane, bool relu) {
  const int n     = ncol * 16 + (lane & 15);
  const int mbase = (lane >> 4) << 3;
#pragma unroll
  for (int r = 0; r < 8; ++r) {
    float v = acc[r];
    if (relu) v = fmaxf(v, 0.0f);
    lds[(mbase + r) * HD + n] = v;
  }
}

__device__ __forceinline__ void store_tile_global(v8f acc, float* __restrict__ out,
                                                  int rowbase, int ncol, int lane,
                                                  bool relu) {
  const int n     = ncol * 16 + (lane & 15);
  const int mbase = (lane >> 4) << 3;
#pragma unroll
  for (int r = 0; r < 8; ++r) {
    float v = acc[r];
    if (relu) v = fmaxf(v, 0.0f);
    out[(size_t)(rowbase + mbase + r) * HD + n] = v;
  }
}

__device__ __forceinline__ void atomicAddF(float* p, float v) {
  __hip_atomic_fetch_add(p, v, __ATOMIC_RELAXED, __HIP_MEMORY_SCOPE_AGENT);
}

// ---------------------------------------------------------------------------
// Edge MLP: e = relu(relu(edge_attr @ We1 + be1) @ We2 + be2)   [E, HD]
// One wave per 16-edge tile; intermediate staged via per-wave LDS region.
// ---------------------------------------------------------------------------
__global__ __launch_bounds__(WPB * WAVE) void edge_mlp_kernel(
    const float* __restrict__ edge_attr,
    const float* __restrict__ We1, const float* __restrict__ be1,
    const float* __restrict__ We2, const float* __restrict__ be2,
    float* __restrict__ e_out) {
  __shared__ float t1buf[WPB][16 * HD];
  const int lane  = threadIdx.x & (WAVE - 1);
  const int wid   = threadIdx.x / WAVE;
  const int tile  = blockIdx.x * WPB + wid;
  const bool valid = tile < (NEDGES / 16);
  const int ebase = tile * 16;

  if (valid) {
    const float* A = edge_attr + (size_t)ebase * EIN;
#pragma unroll
    for (int nc = 0; nc < HD / 16; ++nc) {
      v8f acc = gemm16(A, EIN, We1, be1, nc, EIN, lane);
      store_tile_lds(acc, t1buf[wid], nc, lane, true);
    }
  }
  __syncthreads();
  if (valid) {
#pragma unroll
    for (int nc = 0; nc < HD / 16; ++nc) {
      v8f acc = gemm16(t1buf[wid], HD, We2, be2, nc, HD, lane);
      store_tile_global(acc, e_out, ebase, nc, lane, true);
    }
  }
}

// ---------------------------------------------------------------------------
// Message + scatter: aggr[dst] += relu(h[src] + e)   (one wave per edge)
// ---------------------------------------------------------------------------
__global__ __launch_bounds__(256) void msg_scatter_kernel(
    const float* __restrict__ h, const float* __restrict__ e,
    const int* __restrict__ src, const int* __restrict__ dst,
    float* __restrict__ aggr) {
  const int lane = threadIdx.x & (WAVE - 1);
  const int eid  = (int)((blockIdx.x * (unsigned)blockDim.x + threadIdx.x) / WAVE);
  if (eid >= NEDGES) return;
  const int s = src[eid];
  const int d = dst[eid];
  const float4* hp = (const float4*)(h + (size_t)s * HD);
  const float4* ep = (const float4*)(e + (size_t)eid * HD);
  float* ap = aggr + (size_t)d * HD;
  float4 hv = hp[lane];
  float4 ev = ep[lane];
  const int c = lane * 4;
  atomicAddF(ap + c + 0, fmaxf(hv.x + ev.x, 0.0f));
  atomicAddF(ap + c + 1, fmaxf(hv.y + ev.y, 0.0f));
  atomicAddF(ap + c + 2, fmaxf(hv.z + ev.z, 0.0f));
  atomicAddF(ap + c + 3, fmaxf(hv.w + ev.w, 0.0f));
}

// ---------------------------------------------------------------------------
// Node update + MLP: h_out = relu( relu(z@Wa+ba) @ Wb + bb ),
//                    z = (1+eps)*h_in + aggr.   One wave per 16-node tile.
// ---------------------------------------------------------------------------
__global__ __launch_bounds__(WPB * WAVE) void node_mlp_kernel(
    const float* __restrict__ h_in, const float* __restrict__ aggr,
    const float* __restrict__ eps,
    const float* __restrict__ Wa, const float* __restrict__ ba,
    const float* __restrict__ Wb, const float* __restrict__ bb,
    float* __restrict__ h_out) {
  __shared__ float zbuf[WPB][16 * HD];
  __shared__ float t1buf[WPB][16 * HD];
  const int lane  = threadIdx.x & (WAVE - 1);
  const int wid   = threadIdx.x / WAVE;
  const int tile  = blockIdx.x * WPB + wid;
  const bool valid = tile < (NNODES / 16);
  const int nbase = tile * 16;
  const float ep  = 1.0f + eps[0];

  if (valid) {
    const float4* hp = (const float4*)(h_in + (size_t)nbase * HD);
    const float4* ap = (const float4*)(aggr + (size_t)nbase * HD);
    float4* zp = (float4*)zbuf[wid];
#pragma unroll
    for (int i = lane; i < 16 * HD / 4; i += WAVE) {
      float4 hv = hp[i];
      float4 av = ap[i];
      float4 z;
      z.x = fmaf(ep, hv.x, av.x);
      z.y = fmaf(ep, hv.y, av.y);
      z.z = fmaf(ep, hv.z, av.z);
      z.w = fmaf(ep, hv.w, av.w);
      zp[i] = z;
    }
  }
  __syncthreads();
  if (valid) {
#pragma unroll
    for (int nc = 0; nc < HD / 16; ++nc) {
      v8f acc = gemm16(zbuf[wid], HD, Wa, ba, nc, HD, lane);
      store_tile_lds(acc, t1buf[wid], nc, lane, true);
    }
  }
  __syncthreads();
  if (valid) {
#pragma unroll
    for (int nc = 0; nc < HD / 16; ++nc) {
      v8f acc = gemm16(t1buf[wid], HD, Wb, bb, nc, HD, lane);
      store_tile_global(acc, h_out, nbase, nc, lane, true);
    }
  }
}

// ---------------------------------------------------------------------------
// Global add-pool: hg[batch[n]] += h[n]   (one wave per node)
// ---------------------------------------------------------------------------
__global__ __launch_bounds__(256) void pool_kernel(
    const float* __restrict__ h, const int* __restrict__ batch,
    float* __restrict__ hg) {
  const int lane = threadIdx.x & (WAVE - 1);
  const int nid  = (int)((blockIdx.x * (unsigned)blockDim.x + threadIdx.x) / WAVE);
  if (nid >= NNODES) return;
  const int g = batch[nid];
  const float4* hp = (const float4*)(h + (size_t)nid * HD);
  float* gp = hg + (size_t)g * HD;
  float4 hv = hp[lane];
  const int c = lane * 4;
  atomicAddF(gp + c + 0, hv.x);
  atomicAddF(gp + c + 1, hv.y);
  atomicAddF(gp + c + 2, hv.z);
  atomicAddF(gp + c + 3, hv.w);
}

// ---------------------------------------------------------------------------
// Final projection: out = relu(hg @ Wp + bp).  hg is row-major in global, so
// A operands load directly in A-layout (no LDS staging needed).
// ---------------------------------------------------------------------------
__global__ __launch_bounds__(WPB * WAVE) void proj_kernel(
    const float* __restrict__ hg, const float* __restrict__ Wp,
    const float* __restrict__ bp, float* __restrict__ out) {
  const int lane = threadIdx.x & (WAVE - 1);
  const int wid  = threadIdx.x / WAVE;
  const int tile = blockIdx.x * WPB + wid;
  if (tile >= NGRAPH / 16) return;
  const int rbase = tile * 16;
  const float* A = hg + (size_t)rbase * HD;
#pragma unroll
  for (int nc = 0; nc < HD / 16; ++nc) {
    v8f acc = gemm16(A, HD, Wp, bp, nc, HD, lane);
    store_tile_global(acc, out, rbase, nc, lane, true);
  }
}

// ---------------------------------------------------------------------------
// Host launcher
// ---------------------------------------------------------------------------
extern "C" void kernel_launch(void* const* d_in, const int* in_sizes, int n_in,
                              void* d_out, int out_size, void* d_ws, size_t ws_size,
                              hipStream_t stream) {
  (void)in_sizes; (void)n_in; (void)out_size; (void)ws_size;

  const float* x         = (const float*)d_in[0];
  const int*   edge_idx  = (const int*)d_in[1];
  const float* edge_attr = (const float*)d_in[2];
  const int*   batch     = (const int*)d_in[3];
  const float* We1 = (const float*)d_in[4];
  const float* be1 = (const float*)d_in[5];
  const float* We2 = (const float*)d_in[6];
  const float* be2 = (const float*)d_in[7];
  const float* Wp  = (const float*)d_in[8];
  const float* bp  = (const float*)d_in[9];
  const float *Wa[3], *ba[3], *Wb[3], *bb[3], *eps[3];
  for (int l = 0; l < 3; ++l) {
    Wa[l]  = (const float*)d_in[10 + 5 * l];
    ba[l]  = (const float*)d_in[11 + 5 * l];
    Wb[l]  = (const float*)d_in[12 + 5 * l];
    bb[l]  = (const float*)d_in[13 + 5 * l];
    eps[l] = (const float*)d_in[14 + 5 * l];
  }

  // Workspace layout (fp32): e[E,HD] | h0[N,HD] | h1[N,HD] | aggr[N,HD] | hg[G,HD]
  char* ws = (char*)d_ws;
  float* e    = (float*)ws; ws += (size_t)NEDGES * HD * sizeof(float);
  float* h0   = (float*)ws; ws += (size_t)NNODES * HD * sizeof(float);
  float* h1   = (float*)ws; ws += (size_t)NNODES * HD * sizeof(float);
  float* aggr = (float*)ws; ws += (size_t)NNODES * HD * sizeof(float);
  float* hg   = (float*)ws;

  const int* srcI = edge_idx;           // edge_index[0, :]
  const int* dstI = edge_idx + NEDGES;  // edge_index[1, :]

  // 1) Edge MLP
  const int etiles = NEDGES / 16;
  edge_mlp_kernel<<<(etiles + WPB - 1) / WPB, WPB * WAVE, 0, stream>>>(
      edge_attr, We1, be1, We2, be2, e);

  // 2) Three GINE layers
  const int ntiles = NNODES / 16;
  const int scatter_blocks = (NEDGES * WAVE + 255) / 256;
  const float* hcur = x;
  float* houts[3] = {h0, h1, h0};
  for (int l = 0; l < 3; ++l) {
    hipMemsetAsync(aggr, 0, (size_t)NNODES * HD * sizeof(float), stream);
    msg_scatter_kernel<<<scatter_blocks, 256, 0, stream>>>(hcur, e, srcI, dstI, aggr);
    node_mlp_kernel<<<(ntiles + WPB - 1) / WPB, WPB * WAVE, 0, stream>>>(
        hcur, aggr, eps[l], Wa[l], ba[l], Wb[l], bb[l], houts[l]);
    hcur = houts[l];
  }

  // 3) Pool + projection
  hipMemsetAsync(hg, 0, (size_t)NGRAPH * HD * sizeof(float), stream);
  const int pool_blocks = (NNODES * WAVE + 255) / 256;
  pool_kernel<<<pool_blocks, 256, 0, stream>>>(hcur, batch, hg);
  proj_kernel<<<(NGRAPH / 16 + WPB - 1) / WPB, WPB * WAVE, 0, stream>>>(
      hg, Wp, bp, (float*)d_out);
}